// CornerNet_Loss_89352499626524
// MI455X (gfx1250) — compile-verified
//
#include <hip/hip_runtime.h>

// CornerNet loss for MI455X (gfx1250, wave32).
// Strategy: memory-bound focal loss streamed with NT 128-bit loads + prefetch,
// per-wave partial reduction via V_WMMA_F32_16X16X4_F32, deterministic
// two-stage reduction, tiny AE/offset losses in a single finalize block.

#define EPSF 1.0e-4f

typedef float v2f __attribute__((ext_vector_type(2)));
typedef float v4f __attribute__((ext_vector_type(4)));
typedef float v8f __attribute__((ext_vector_type(8)));

constexpr int S_ = 2, B_ = 16, C_ = 80, H_ = 128, W_ = 128, N_ = 128;
constexpr int HW_   = H_ * W_;             // 16384
constexpr int BCHW_ = B_ * C_ * HW_;       // 20,971,520
constexpr int NV4_  = BCHW_ / 4;           // 5,242,880 float4s
constexpr int FBLK_ = 512;                 // focal blocks per (stack,corner) combo

// ---------------------------------------------------------------------------
// Kernel A: focal-loss partial sums.
//   grid = (FBLK_, 4): blockIdx.y = combo (s = y>>1, corner = y&1)
//   Each block writes {pos_loss, neg_loss, num_pos} partials to ws.
// ---------------------------------------------------------------------------
__global__ __launch_bounds__(256)
void focal_partial_kernel(const float* __restrict__ tl_heats,
                          const float* __restrict__ br_heats,
                          const float* __restrict__ gt_tl,
                          const float* __restrict__ gt_br,
                          float* __restrict__ part)
{
    const int combo  = blockIdx.y;     // 0..3
    const int s      = combo >> 1;
    const int corner = combo & 1;

    const float* heat = (corner ? br_heats : tl_heats) + (size_t)s * BCHW_;
    const float* gt   = corner ? gt_br : gt_tl;
    const v4f* hp = (const v4f*)heat;
    const v4f* gp = (const v4f*)gt;

    float pos = 0.0f, neg = 0.0f, np = 0.0f;
    const int stride = gridDim.x * blockDim.x;
    for (int i = blockIdx.x * blockDim.x + threadIdx.x; i < NV4_; i += stride) {
        // predictions: read-once -> non-temporal (keep GT maps hot in 192MB L2)
        v4f x = __builtin_nontemporal_load(hp + i);
        v4f g = gp[i];
        __builtin_prefetch(hp + i + stride, 0, 1);   // global_prefetch_b8
#pragma unroll
        for (int k = 0; k < 4; ++k) {
            float p = 1.0f / (1.0f + expf(-x[k]));
            p = fminf(fmaxf(p, EPSF), 1.0f - EPSF);
            const float gv  = g[k];
            const float omp = 1.0f - p;
            float w = 1.0f - gv; w *= w; w *= w;          // (1-g)^4
            const float plv = logf(p)   * omp * omp;      // pos term
            const float nlv = logf(omp) * p * p * w;      // neg term
            const bool isPos = (gv == 1.0f);
            pos += isPos ? plv : 0.0f;
            neg += isPos ? 0.0f : nlv;
            np  += isPos ? 1.0f : 0.0f;
        }
    }

    // ---- wave-level reduce of (pos, neg) via WMMA f32 16x16x4 ----
    // A (16x4): lane m<16 -> A[m][0]=pos, A[m][1]=neg; lane m+16 -> A[m][2],A[m][3].
    // B (4x16) selector: col0 = rows {0,2}, col1 = rows {1,3}
    //   => D[m][0] = pos_m + pos_{m+16},  D[m][1] = neg_m + neg_{m+16}
    const unsigned lane = threadIdx.x & 31u;
    v2f a; a[0] = pos; a[1] = neg;
    v2f b;
    b[0] = (lane == 0u || lane == 16u) ? 1.0f : 0.0f;
    b[1] = (lane == 1u || lane == 17u) ? 1.0f : 0.0f;
    v8f c = {0.f, 0.f, 0.f, 0.f, 0.f, 0.f, 0.f, 0.f};
    v8f d = __builtin_amdgcn_wmma_f32_16x16x4_f32(false, a, false, b,
                                                  (short)0, c, false, false);
    // lane n<16 holds D[0..7][n]; lane n>=16 holds D[8..15][n-16]
    float t = d[0] + d[1] + d[2] + d[3] + d[4] + d[5] + d[6] + d[7];
    float u = t + __shfl_xor(t, 16, 32);
    const float posT = __shfl(u, 0, 32);   // full 32-lane pos sum
    const float negT = __shfl(u, 1, 32);   // full 32-lane neg sum

    float npT = np;                        // num_pos: plain shuffle tree
#pragma unroll
    for (int off = 16; off > 0; off >>= 1) npT += __shfl_xor(npT, off, 32);

    __shared__ float sw[8][3];
    const int wv = threadIdx.x >> 5;
    if (lane == 0) { sw[wv][0] = posT; sw[wv][1] = negT; sw[wv][2] = npT; }
    __syncthreads();
    if (threadIdx.x == 0) {
        float P = 0.f, Ng = 0.f, Np = 0.f;
        for (int w8 = 0; w8 < 8; ++w8) { P += sw[w8][0]; Ng += sw[w8][1]; Np += sw[w8][2]; }
        const int slot = combo * gridDim.x + blockIdx.x;
        part[slot * 3 + 0] = P;
        part[slot * 3 + 1] = Ng;
        part[slot * 3 + 2] = Np;
    }
}

// ---------------------------------------------------------------------------
// Block-wide sum (deterministic), result broadcast to all threads.
// ---------------------------------------------------------------------------
__device__ float block_reduce(float v, float* red)
{
    const int t = threadIdx.x;
    __syncthreads();
    red[t] = v;
    __syncthreads();
    for (int s2 = 128; s2 > 0; s2 >>= 1) {
        if (t < s2) red[t] += red[t + s2];
        __syncthreads();
    }
    const float r = red[0];
    __syncthreads();
    return r;
}

// ---------------------------------------------------------------------------
// Kernel B: reduce focal partials + AE (pull/push) + offset losses -> scalar.
// Single block of 256 threads; total extra work ~0.6M cheap iterations.
// ---------------------------------------------------------------------------
__global__ __launch_bounds__(256)
void finalize_kernel(const float* __restrict__ part, int nblk,
                     const float* __restrict__ tl_tags,
                     const float* __restrict__ br_tags,
                     const float* __restrict__ tl_offs,
                     const float* __restrict__ br_offs,
                     const unsigned char* __restrict__ tag_mask,
                     const unsigned char* __restrict__ off_tl_mask,
                     const unsigned char* __restrict__ off_br_mask,
                     const float* __restrict__ gt_tl_off,
                     const float* __restrict__ gt_br_off,
                     const int* __restrict__ tl_tag_ind,
                     const int* __restrict__ br_tag_ind,
                     const int* __restrict__ tl_off_ind,
                     const int* __restrict__ br_off_ind,
                     float* __restrict__ out)
{
    __shared__ float red[256];
    __shared__ float s_t0[B_ * N_], s_t1[B_ * N_], s_mean[B_ * N_], s_m[B_ * N_];
    __shared__ float s_num[B_];
    const int tid = threadIdx.x;

    // ---- focal: reduce per-block partials for each of the 4 combos ----
    float fsum = 0.0f;
    for (int c = 0; c < 4; ++c) {
        float pos = 0.f, neg = 0.f, np = 0.f;
        for (int i = tid; i < nblk; i += blockDim.x) {
            const int slot = c * nblk + i;
            pos += part[slot * 3 + 0];
            neg += part[slot * 3 + 1];
            np  += part[slot * 3 + 2];
        }
        pos = block_reduce(pos, red);
        neg = block_reduce(neg, red);
        np  = block_reduce(np,  red);
        fsum += (np > 0.0f) ? (-(pos + neg) / np) : (-neg);
    }

    // ---- associative embedding (pull/push), per stack ----
    float pull = 0.0f, push = 0.0f;
    for (int s = 0; s < S_; ++s) {
        const float* t0p = tl_tags + (size_t)s * B_ * HW_;
        const float* t1p = br_tags + (size_t)s * B_ * HW_;
        __syncthreads();
        for (int i = tid; i < B_ * N_; i += blockDim.x) {
            const int b = i >> 7;                       // N_ = 128
            const float a0 = t0p[b * HW_ + tl_tag_ind[i]];
            const float a1 = t1p[b * HW_ + br_tag_ind[i]];
            s_t0[i] = a0; s_t1[i] = a1;
            s_mean[i] = 0.5f * (a0 + a1);
            s_m[i] = tag_mask[i] ? 1.0f : 0.0f;
        }
        __syncthreads();
        if (tid < B_) {
            float nm = 0.f;
            for (int n = 0; n < N_; ++n) nm += s_m[tid * N_ + n];
            s_num[tid] = nm;
        }
        __syncthreads();

        float pp = 0.0f;
        for (int i = tid; i < B_ * N_; i += blockDim.x) {
            const int b = i >> 7;
            const float inv = 1.0f / (s_num[b] + EPSF);
            const float d0 = s_t0[i] - s_mean[i];
            const float d1 = s_t1[i] - s_mean[i];
            pp += s_m[i] * d0 * d0 * inv + s_m[i] * d1 * d1 * inv;
        }
        pull += block_reduce(pp, red);

        float ps = 0.0f;
        for (int idx = tid; idx < B_ * N_ * N_; idx += blockDim.x) {
            const int b   = idx >> 14;                  // N*N = 16384
            const int rem = idx & (N_ * N_ - 1);
            const int i2  = rem >> 7;
            const int j   = rem & (N_ - 1);
            const float numc = s_num[b];
            const float num2 = (numc - 1.0f) * numc;
            const float dmn  = fabsf(s_mean[b * N_ + j] - s_mean[b * N_ + i2]);
            float dist = fmaxf(1.0f - dmn, 0.0f);
            dist = (dist - 1.0f / (numc + EPSF)) / (num2 + EPSF);
            ps += s_m[b * N_ + i2] * s_m[b * N_ + j] * dist;
        }
        push += block_reduce(ps, red);
    }

    // ---- offset losses (smooth L1), per stack x corner ----
    float off = 0.0f;
    for (int s = 0; s < S_; ++s) {
        for (int corner = 0; corner < 2; ++corner) {
            const float* offs = (corner ? br_offs : tl_offs) + (size_t)s * B_ * 2 * HW_;
            const unsigned char* msk = corner ? off_br_mask : off_tl_mask;
            const int*   ind = corner ? br_off_ind : tl_off_ind;
            const float* gto = corner ? gt_br_off  : gt_tl_off;
            float sl = 0.f, nm = 0.f;
            for (int i = tid; i < B_ * N_; i += blockDim.x) {
                const int b = i >> 7;
                const float m = msk[i] ? 1.0f : 0.0f;
                nm += m;
                const int id = ind[i];
                const float* base = offs + (size_t)b * 2 * HW_;
                const float o0 = base[id];
                const float o1 = base[HW_ + id];
                const float d0 = fabsf(o0 - gto[i * 2 + 0]);
                const float d1 = fabsf(o1 - gto[i * 2 + 1]);
                const float l0 = (d0 < 1.0f) ? 0.5f * d0 * d0 : d0 - 0.5f;
                const float l1 = (d1 < 1.0f) ? 0.5f * d1 * d1 : d1 - 0.5f;
                sl += m * (l0 + l1);
            }
            sl = block_reduce(sl, red);
            nm = block_reduce(nm, red);
            off += sl / (nm + EPSF);
        }
    }

    if (tid == 0) out[0] = (fsum + pull + push + off) / (float)S_;
}

// ---------------------------------------------------------------------------
extern "C" void kernel_launch(void* const* d_in, const int* in_sizes, int n_in,
                              void* d_out, int out_size, void* d_ws, size_t ws_size,
                              hipStream_t stream)
{
    (void)in_sizes; (void)n_in; (void)out_size; (void)ws_size;
    const float* tl_heats = (const float*)d_in[0];
    const float* br_heats = (const float*)d_in[1];
    const float* tl_tags  = (const float*)d_in[2];
    const float* br_tags  = (const float*)d_in[3];
    const float* tl_offs  = (const float*)d_in[4];
    const float* br_offs  = (const float*)d_in[5];
    const float* gt_tl    = (const float*)d_in[6];
    const float* gt_br    = (const float*)d_in[7];
    const unsigned char* tag_mask    = (const unsigned char*)d_in[8];
    const unsigned char* off_tl_mask = (const unsigned char*)d_in[9];
    const unsigned char* off_br_mask = (const unsigned char*)d_in[10];
    const float* gt_tl_off = (const float*)d_in[11];
    const float* gt_br_off = (const float*)d_in[12];
    const int* tl_tag_ind  = (const int*)d_in[13];
    const int* br_tag_ind  = (const int*)d_in[14];
    const int* tl_off_ind  = (const int*)d_in[15];
    const int* br_off_ind  = (const int*)d_in[16];

    float* part = (float*)d_ws;   // 4 * FBLK_ * 3 floats = 24 KB

    dim3 gridA(FBLK_, 4, 1);
    focal_partial_kernel<<<gridA, 256, 0, stream>>>(tl_heats, br_heats, gt_tl, gt_br, part);
    finalize_kernel<<<1, 256, 0, stream>>>(part, FBLK_,
                                           tl_tags, br_tags, tl_offs, br_offs,
                                           tag_mask, off_tl_mask, off_br_mask,
                                           gt_tl_off, gt_br_off,
                                           tl_tag_ind, br_tag_ind,
                                           tl_off_ind, br_off_ind,
                                           (float*)d_out);
}